// CenterTOpEXnewMultiC_80788334837912
// MI455X (gfx1250) — compile-verified
//
#include <hip/hip_runtime.h>
#include <hip/hip_bf16.h>

typedef __attribute__((ext_vector_type(2))) float v2f;
typedef __attribute__((ext_vector_type(8))) float v8f;

#define NPIX 65536   // 256*256
#define NCH  128
#define NBB  8

// ---------------- workspace zero (graph-capture safe) ----------------
__global__ void ws_zero_kernel(float* ws) {
    int t = blockIdx.x * 256 + threadIdx.x;
    if (t < 257) ws[t] = 0.0f;
}

// ---------------- pass 1: dists / labels / onehot via WMMA ----------------
// grid: (NPIX/256, NBB), block 256 (8 waves). Each wave: 2 tiles of 16 pixels.
// D = A(16x4) x B(4x16) + C, A rows 0,1 = centers, B columns = pixels.
__global__ void __launch_bounds__(256)
k_assign(const float* __restrict__ F, const float* __restrict__ centers,
         float* __restrict__ out) {
    __shared__ float cLDS[2 * NCH];
    int t = threadIdx.x;
    cLDS[t] = centers[t];
    __syncthreads();

    const int wave = t >> 5;
    const int lane = t & 31;
    const int h    = lane >> 4;     // K half: 0 -> K{0,1}, 1 -> K{2,3}
    const int idx  = lane & 15;     // A row (center id) / B column (pixel)
    const int b    = blockIdx.y;
    const int p0   = blockIdx.x * 256 + wave * 32;   // 2 tiles of 16 pixels

    const float* Fb = F + (size_t)b * NCH * NPIX;
    const int   row = idx & 1;            // which center row this lane mirrors
    const bool  live = (idx < 2);         // A rows >=2 are zero

    v8f acc0 = {};
    v8f acc1 = {};

#pragma unroll
    for (int kb = 0; kb < NCH; kb += 4) {
        const int kOff = kb + 2 * h;
        // A operand: lanes hold center[row][kOff], center[row][kOff+1] (zero if row>=2)
        float2 cv = *(const float2*)(&cLDS[row * NCH + kOff]);
        v2f a;
        a.x = live ? cv.x : 0.0f;
        a.y = live ? cv.y : 0.0f;
        // B operand tile 0: pixel p0+idx, channels kOff, kOff+1 (coalesced per 16-lane half)
        const size_t base = (size_t)kOff * NPIX + (size_t)p0 + idx;
        v2f b0, b1;
        b0.x = Fb[base];
        b0.y = Fb[base + NPIX];
        b1.x = Fb[base + 16];
        b1.y = Fb[base + NPIX + 16];
        acc0 = __builtin_amdgcn_wmma_f32_16x16x4_f32(false, a, false, b0, (short)0, acc0, false, false);
        acc1 = __builtin_amdgcn_wmma_f32_16x16x4_f32(false, a, false, b1, (short)0, acc1, false, false);
    }

    // D layout: acc[0] lanes 0-15 = row M=0 (center0) for pixel n=lane,
    //           acc[1] lanes 0-15 = row M=1 (center1). Lanes 16-31: rows 8,9 (zero).
    float* OUT1 = out + 256;                          // labels        [bb*N]
    float* OUT2 = OUT1 + (size_t)NBB * NPIX;          // onehot        [bb*2N]
    float* OUT3 = OUT2 + (size_t)NBB * 2 * NPIX;      // dist2center   [bb*2N]
    float* OUT4 = OUT3 + (size_t)NBB * 2 * NPIX;      // labelTinit    [bb*N]

    if (lane < 16) {
#pragma unroll
        for (int tile = 0; tile < 2; ++tile) {
            const v8f& acc = tile ? acc1 : acc0;
            const int p = p0 + tile * 16 + lane;
            float d0 = 0.5f * (1.0f - acc[0]);
            float d1 = 0.5f * (1.0f - acc[1]);
            float labf = (d1 < d0) ? 1.0f : 0.0f;   // argmin, first index wins ties
            size_t bi = (size_t)b * NPIX + p;
            OUT1[bi] = labf;
            OUT4[bi] = labf;
            size_t b2 = (size_t)b * (2 * NPIX) + 2 * (size_t)p;
            *(float2*)(&OUT2[b2]) = make_float2(1.0f - labf, labf);
            *(float2*)(&OUT3[b2]) = make_float2(d0, d1);
        }
    }
}

// ---------------- pass 2: masked channel sums over image 7 ----------------
// grid 256 blocks x 256 threads; block owns a 256-pixel tile, loops channels.
__global__ void __launch_bounds__(256)
k_centersum(const float* __restrict__ F, const float* __restrict__ labels7,
            float* __restrict__ ws) {
    __shared__ float acc[257];
    int t = threadIdx.x;
    acc[t] = 0.0f;
    if (t == 0) acc[256] = 0.0f;
    __syncthreads();

    const int p = blockIdx.x * 256 + t;
    const float* F7 = F + (size_t)7 * NCH * NPIX;
    const float lab = labels7[p];           // 0.0 or 1.0

    // count of label==1 pixels (for Num smoothing)
    float c1 = lab;
    for (int m = 16; m > 0; m >>= 1) c1 += __shfl_xor(c1, m, 32);
    if ((t & 31) == 0) atomicAdd(&acc[256], c1);

    for (int c = 0; c < NCH; ++c) {
        float val = F7[(size_t)c * NPIX + p];
        float v1 = val * lab;
        float v0 = val - v1;
        for (int m = 16; m > 0; m >>= 1) {
            v0 += __shfl_xor(v0, m, 32);
            v1 += __shfl_xor(v1, m, 32);
        }
        if ((t & 31) == 0) {
            atomicAdd(&acc[c], v0);
            atomicAdd(&acc[NCH + c], v1);
        }
    }
    __syncthreads();
    atomicAdd(&ws[t], acc[t]);              // ws[k*128+c]
    if (t == 0) atomicAdd(&ws[256], acc[256]);
}

// ---------------- pass 3: centersIterout ----------------
__global__ void k_finalize(const float* __restrict__ centers,
                           const float* __restrict__ ws,
                           float* __restrict__ out) {
    int t = threadIdx.x;                    // 256 threads: t = k*128 + c
    float cnt1 = ws[256];
    int k = t >> 7;
    float Num = (k ? cnt1 : (65536.0f - cnt1)) + 1.0f;
    float centerIter = ws[t] / Num;
    float ci = centers[t];
    out[t] = ci + 0.001f * (centerIter - ci);
}

extern "C" void kernel_launch(void* const* d_in, const int* in_sizes, int n_in,
                              void* d_out, int out_size, void* d_ws, size_t ws_size,
                              hipStream_t stream) {
    const float* F       = (const float*)d_in[0];   // [8,128,256,256]
    const float* centers = (const float*)d_in[1];   // [2,128]
    float* out = (float*)d_out;
    float* ws  = (float*)d_ws;

    ws_zero_kernel<<<2, 256, 0, stream>>>(ws);

    dim3 g1(NPIX / 256, NBB);
    k_assign<<<g1, 256, 0, stream>>>(F, centers, out);

    const float* labels7 = out + 256 + (size_t)7 * NPIX;
    k_centersum<<<NPIX / 256, 256, 0, stream>>>(F, labels7, ws);

    k_finalize<<<1, 256, 0, stream>>>(centers, ws, out);
}